// MeshGraphUnet_90400471646658
// MI455X (gfx1250) — compile-verified
//
#include <hip/hip_runtime.h>
#include <math.h>

#define NNODES 131072
#define NEDGES 524288
#define HDIM   128
#define KPOOL  (NNODES / 2)
#define LNEPS  1e-5f

typedef float v2f __attribute__((ext_vector_type(2)));
typedef float v8f __attribute__((ext_vector_type(8)));

// ---------------- pooling: score / key ----------------

__global__ void k_pnorm(const float* __restrict__ p, float* __restrict__ inv_norm) {
  __shared__ float red[128];
  int t = threadIdx.x;
  float v = p[t];
  red[t] = v * v;
  __syncthreads();
  for (int off = 64; off > 0; off >>= 1) {
    if (t < off) red[t] += red[t + off];
    __syncthreads();
  }
  if (t == 0) *inv_norm = rsqrtf(red[0]);
}

__global__ void k_score(const float* __restrict__ x, const float* __restrict__ p,
                        const float* __restrict__ inv_norm_p,
                        float* __restrict__ score, float* __restrict__ gate,
                        unsigned* __restrict__ key) {
  int wpb = blockDim.x >> 5;
  int w = blockIdx.x * wpb + (threadIdx.x >> 5);
  int lane = threadIdx.x & 31;
  if (w >= NNODES) return;
  float4 xv = ((const float4*)(x + (size_t)w * HDIM))[lane];
  float4 pv = ((const float4*)p)[lane];
  float d = xv.x * pv.x + xv.y * pv.y + xv.z * pv.z + xv.w * pv.w;
  for (int off = 16; off; off >>= 1) d += __shfl_xor(d, off, 32);
  if (lane == 0) {
    float s = d * (*inv_norm_p);
    score[w] = s;
    gate[w] = tanhf(s);
    unsigned u = __float_as_uint(s);
    u = (u & 0x80000000u) ? ~u : (u | 0x80000000u); // order-preserving map
    key[w] = u;
  }
}

// ---------------- radix select (top-KPOOL threshold) ----------------
// state[0]=prefix  state[1]=remaining  state[2]=eqTaken  state[3]=poolCount
// state[8..263] = 256-bin histogram

__global__ void k_rsinit(unsigned* __restrict__ state) {
  state[0] = 0u;
  state[1] = (unsigned)KPOOL;
  state[2] = 0u;
  state[3] = 0u;
}

__global__ void k_hist(const unsigned* __restrict__ key, unsigned* __restrict__ state, int shift) {
  __shared__ unsigned lh[256];
  unsigned* hist = state + 8;
  for (int i = threadIdx.x; i < 256; i += blockDim.x) lh[i] = 0u;
  __syncthreads();
  unsigned long long pHi = (unsigned long long)state[0] >> (shift + 8);
  for (int i = blockIdx.x * blockDim.x + threadIdx.x; i < NNODES; i += gridDim.x * blockDim.x) {
    unsigned kv = key[i];
    if (((unsigned long long)kv >> (shift + 8)) == pHi)
      atomicAdd(&lh[(kv >> shift) & 0xFFu], 1u);
  }
  __syncthreads();
  for (int i = threadIdx.x; i < 256; i += blockDim.x)
    if (lh[i]) atomicAdd(&hist[i], lh[i]);
}

__global__ void k_select(unsigned* __restrict__ state, int shift) {
  unsigned* hist = state + 8;
  unsigned remaining = state[1];
  unsigned cum = 0;
  int d;
  for (d = 255; d >= 0; --d) {
    unsigned c = hist[d];
    if (cum + c >= remaining) break;
    cum += c;
  }
  if (d < 0) d = 0;
  state[0] |= ((unsigned)d) << shift;
  state[1] = remaining - cum;
  for (int i = 0; i < 256; ++i) hist[i] = 0u; // reset for next pass
}

__global__ void k_compact(const unsigned* __restrict__ key, unsigned* __restrict__ state,
                          int* __restrict__ mask, int* __restrict__ mapping) {
  int i = blockIdx.x * blockDim.x + threadIdx.x;
  if (i >= NNODES) return;
  unsigned T = state[0];
  unsigned eqNeed = state[1];
  unsigned kv = key[i];
  int keep = 0;
  if (kv > T) keep = 1;
  else if (kv == T) keep = (atomicAdd(&state[2], 1u) < eqNeed) ? 1 : 0;
  if (keep) {
    mapping[i] = (int)atomicAdd(&state[3], 1u);
    mask[i] = 1;
  } else {
    mapping[i] = 0;
    mask[i] = 0;
  }
}

// ---------------- message passing (gather + scatter-add) ----------------
// one wave per directed message; lanes cover 128 channels as float4

__global__ void k_mp_down(const float* __restrict__ x, const float* __restrict__ gate,
                          const int* __restrict__ senders, const int* __restrict__ receivers,
                          const int* __restrict__ mask, const int* __restrict__ mapping,
                          float* __restrict__ agg /*[KPOOL,128]*/) {
  int wpb = blockDim.x >> 5;
  long long j = (long long)blockIdx.x * wpb + (threadIdx.x >> 5);
  int lane = threadIdx.x & 31;
  if (j >= 2LL * NEDGES) return;
  int s, r;
  if (j < NEDGES) { s = senders[j]; r = receivers[j]; }
  else            { s = receivers[j - NEDGES]; r = senders[j - NEDGES]; }
  if (!(mask[s] & mask[r])) return;
  float g = gate[s];
  float4 v = ((const float4*)(x + (size_t)s * HDIM))[lane];
  float* dst = agg + (size_t)mapping[r] * HDIM + lane * 4;
  unsafeAtomicAdd(dst + 0, v.x * g);
  unsafeAtomicAdd(dst + 1, v.y * g);
  unsafeAtomicAdd(dst + 2, v.z * g);
  unsafeAtomicAdd(dst + 3, v.w * g);
}

__global__ void k_mp_up(const float* __restrict__ x, const float* __restrict__ xd /*[KPOOL,128]*/,
                        const int* __restrict__ senders, const int* __restrict__ receivers,
                        const int* __restrict__ mask, const int* __restrict__ mapping,
                        float* __restrict__ agg /*[N,256]*/) {
  int wpb = blockDim.x >> 5;
  long long j = (long long)blockIdx.x * wpb + (threadIdx.x >> 5);
  int lane = threadIdx.x & 31;
  if (j >= 2LL * NEDGES) return;
  int s, r;
  if (j < NEDGES) { s = senders[j]; r = receivers[j]; }
  else            { s = receivers[j - NEDGES]; r = senders[j - NEDGES]; }
  float* dst = agg + (size_t)r * (2 * HDIM);
  float4 xv = ((const float4*)(x + (size_t)s * HDIM))[lane];
  float* d1 = dst + HDIM + lane * 4; // skip half
  unsafeAtomicAdd(d1 + 0, xv.x);
  unsafeAtomicAdd(d1 + 1, xv.y);
  unsafeAtomicAdd(d1 + 2, xv.z);
  unsafeAtomicAdd(d1 + 3, xv.w);
  if (mask[s]) { // recovered half
    float4 dv = ((const float4*)(xd + (size_t)mapping[s] * HDIM))[lane];
    float* d0 = dst + lane * 4;
    unsafeAtomicAdd(d0 + 0, dv.x);
    unsafeAtomicAdd(d0 + 1, dv.y);
    unsafeAtomicAdd(d0 + 2, dv.z);
    unsafeAtomicAdd(d0 + 3, dv.w);
  }
}

// ---------------- WMMA f32 GEMM: D[M,128] = act(A[M,K] @ B[K,128] + bias) ----------------
// One wave per 16-row M block computing ALL 8 N-tiles: the A fragment is
// loaded once per K-step and fed to 8 independent V_WMMA_F32_16X16X4_F32
// (distinct accumulators -> no D->A/B hazard, they pipeline back-to-back).
// This cuts streamed-A traffic 8x vs tile-per-wave.
// A frag (16x4): lane&15 = row; VGPR pair = (K+2*(lane>>4), +1).
// B frag (4x16): lane&15 = col; same K pairing. C/D: lane&15 = col, vgpr v -> row v + 8*(lane>>4).

template <int KDIM, bool ACT>
__global__ void k_gemm(const float* __restrict__ A, const float* __restrict__ B,
                       const float* __restrict__ bias, float* __restrict__ D) {
  int mtile = blockIdx.x * blockDim.y + threadIdx.y;
  int lane = threadIdx.x;
  int r16 = lane & 15;
  int hi = lane >> 4;
  const float* Arow = A + ((size_t)mtile * 16 + r16) * KDIM;
  v8f acc[8];
#pragma unroll
  for (int n = 0; n < 8; ++n) acc[n] = (v8f){};

#pragma unroll 2
  for (int kk = 0; kk < KDIM; kk += 4) {
    int ko = kk + 2 * hi;
    v2f a;
    a.x = Arow[ko];
    a.y = Arow[ko + 1];
    const float* B0 = B + (size_t)ko * HDIM + r16;
    const float* B1 = B0 + HDIM;
#pragma unroll
    for (int n = 0; n < 8; ++n) {
      v2f b;
      b.x = B0[n * 16];
      b.y = B1[n * 16];
      acc[n] = __builtin_amdgcn_wmma_f32_16x16x4_f32(false, a, false, b, (short)0, acc[n],
                                                     false, false);
    }
  }

#pragma unroll
  for (int n = 0; n < 8; ++n) {
    int col = n * 16 + r16;
    float bb = bias[col];
#pragma unroll
    for (int v = 0; v < 8; ++v) {
      int row = mtile * 16 + v + 8 * hi;
      float val = acc[n][v] + bb;
      if (ACT) val = 0.5f * val * (1.0f + erff(val * 0.70710678118654752f)); // exact GELU
      D[(size_t)row * HDIM + col] = val;
    }
  }
}

// ---------------- LayerNorm (wave per row of 128) ----------------

__global__ void k_ln(const float* __restrict__ X, const float* __restrict__ g,
                     const float* __restrict__ b, float* __restrict__ Y, int rows) {
  int wpb = blockDim.x >> 5;
  int w = blockIdx.x * wpb + (threadIdx.x >> 5);
  int lane = threadIdx.x & 31;
  if (w >= rows) return;
  float4 v = ((const float4*)(X + (size_t)w * HDIM))[lane];
  float s = v.x + v.y + v.z + v.w;
  float q = v.x * v.x + v.y * v.y + v.z * v.z + v.w * v.w;
  for (int off = 16; off; off >>= 1) {
    s += __shfl_xor(s, off, 32);
    q += __shfl_xor(q, off, 32);
  }
  float mu = s * (1.0f / HDIM);
  float var = q * (1.0f / HDIM) - mu * mu;
  float rs = rsqrtf(var + LNEPS);
  float4 gv = ((const float4*)g)[lane];
  float4 bv = ((const float4*)b)[lane];
  float4 o;
  o.x = (v.x - mu) * rs * gv.x + bv.x;
  o.y = (v.y - mu) * rs * gv.y + bv.y;
  o.z = (v.z - mu) * rs * gv.z + bv.z;
  o.w = (v.w - mu) * rs * gv.w + bv.w;
  ((float4*)(Y + (size_t)w * HDIM))[lane] = o;
}

// ---------------- host launch ----------------

extern "C" void kernel_launch(void* const* d_in, const int* in_sizes, int n_in,
                              void* d_out, int out_size, void* d_ws, size_t ws_size,
                              hipStream_t stream) {
  const float* x       = (const float*)d_in[0];
  const int*   ei      = (const int*)d_in[1];
  const float* pool_p  = (const float*)d_in[2];
  const float* dW1 = (const float*)d_in[3];
  const float* db1 = (const float*)d_in[4];
  const float* dW2 = (const float*)d_in[5];
  const float* db2 = (const float*)d_in[6];
  const float* dW3 = (const float*)d_in[7];
  const float* db3 = (const float*)d_in[8];
  const float* dg  = (const float*)d_in[9];
  const float* dbe = (const float*)d_in[10];
  const float* uW1 = (const float*)d_in[11];
  const float* ub1 = (const float*)d_in[12];
  const float* uW2 = (const float*)d_in[13];
  const float* ub2 = (const float*)d_in[14];
  const float* uW3 = (const float*)d_in[15];
  const float* ub3 = (const float*)d_in[16];
  const float* ug  = (const float*)d_in[17];
  const float* ube = (const float*)d_in[18];
  const int* senders = ei;
  const int* receivers = ei + NEDGES;
  float* out = (float*)d_out;

  // workspace carve-out
  char* ws = (char*)d_ws;
  size_t off = 0;
  auto carve = [&](size_t bytes) -> char* {
    char* p = ws + off;
    off += (bytes + 255) & ~(size_t)255;
    return p;
  };
  unsigned* state   = (unsigned*)carve(4096);
  float*    invnp   = (float*)carve(256);
  float*    score   = (float*)carve((size_t)NNODES * 4);
  float*    gate    = (float*)carve((size_t)NNODES * 4);
  unsigned* key     = (unsigned*)carve((size_t)NNODES * 4);
  int*      mask    = (int*)carve((size_t)NNODES * 4);
  int*      mapping = (int*)carve((size_t)NNODES * 4);
  float*    agg_d   = (float*)carve((size_t)KPOOL * HDIM * 4);
  float*    hA      = (float*)carve((size_t)KPOOL * HDIM * 4);
  float*    hB      = (float*)carve((size_t)KPOOL * HDIM * 4);
  float*    agg_u   = (float*)carve((size_t)NNODES * 2 * HDIM * 4);
  float*    uA      = (float*)carve((size_t)NNODES * HDIM * 4);
  float*    uB      = (float*)carve((size_t)NNODES * HDIM * 4);

  // accumulators / select state must be zero every call
  hipMemsetAsync(state, 0, 4096, stream);
  hipMemsetAsync(agg_d, 0, (size_t)KPOOL * HDIM * 4, stream);
  hipMemsetAsync(agg_u, 0, (size_t)NNODES * 2 * HDIM * 4, stream);

  k_rsinit<<<1, 1, 0, stream>>>(state);
  k_pnorm<<<1, 128, 0, stream>>>(pool_p, invnp);
  k_score<<<NNODES / 8, 256, 0, stream>>>(x, pool_p, invnp, score, gate, key);

  for (int p = 0; p < 4; ++p) {
    int shift = 24 - 8 * p;
    k_hist<<<512, 256, 0, stream>>>(key, state, shift);
    k_select<<<1, 1, 0, stream>>>(state, shift);
  }
  k_compact<<<NNODES / 256, 256, 0, stream>>>(key, state, mask, mapping);

  // down: message passing then MLP + LN
  k_mp_down<<<(2 * NEDGES) / 8, 256, 0, stream>>>(x, gate, senders, receivers, mask, mapping, agg_d);
  dim3 gblk(32, 8); // 8 waves/block, one 16-row M block per wave
  k_gemm<128, true ><<<KPOOL / 16 / 8, gblk, 0, stream>>>(agg_d, dW1, db1, hA);
  k_gemm<128, true ><<<KPOOL / 16 / 8, gblk, 0, stream>>>(hA, dW2, db2, hB);
  k_gemm<128, false><<<KPOOL / 16 / 8, gblk, 0, stream>>>(hB, dW3, db3, hA);
  k_ln<<<KPOOL / 8, 256, 0, stream>>>(hA, dg, dbe, hB, KPOOL); // hB = x_d

  // up: unpool+concat fused into message passing, MLP + LN
  k_mp_up<<<(2 * NEDGES) / 8, 256, 0, stream>>>(x, hB, senders, receivers, mask, mapping, agg_u);
  k_gemm<256, true ><<<NNODES / 16 / 8, gblk, 0, stream>>>(agg_u, uW1, ub1, uA);
  k_gemm<128, true ><<<NNODES / 16 / 8, gblk, 0, stream>>>(uA, uW2, ub2, uB);
  k_gemm<128, false><<<NNODES / 16 / 8, gblk, 0, stream>>>(uB, uW3, ub3, uA);
  k_ln<<<NNODES / 8, 256, 0, stream>>>(uA, ug, ube, out, NNODES);
}